// UniGCNIILayer_27230092657701
// MI455X (gfx1250) — compile-verified
//
#include <hip/hip_runtime.h>

#define NN 16384   // nodes
#define NE 16384   // edges
#define C  64
#define WAUG 80    // 64 channels + [1.0, deg] + pad to 5 WMMA tiles
#define KT 32      // K slab per iteration
#define KTP 33     // padded K stride in transposed LDS slab

typedef __attribute__((ext_vector_type(2))) float v2f;
typedef __attribute__((ext_vector_type(8))) float v8f;

__device__ __forceinline__ v8f wmma4(v2f a, v2f b, v8f c) {
  // D = A(16x4) * B(4x16) + C, fp32
  return __builtin_amdgcn_wmma_f32_16x16x4_f32(false, a, false, b, (short)0, c,
                                               false, false);
}

// async copy one dword from global memory at byte offset g_off past sbase to the LDS byte address lds_off   (ASYNCcnt)
__device__ __forceinline__ void async_ld_b32(uint32_t lds_off, uint32_t g_off,
                                             const float* sbase) {
  asm volatile("global_load_async_to_lds_b32 %0, %1, %2"
               :: "v"(lds_off), "v"(g_off), "s"(sbase)
               : "memory");
}

template <int N>
__device__ __forceinline__ void wait_asynccnt() {
  asm volatile("s_wait_asynccnt %0" :: "n"(N) : "memory");
}

// ---------------- node degree: row sums of B ----------------
__global__ __launch_bounds__(256) void k_degree(const float* __restrict__ B,
                                                float* __restrict__ deg) {
  __shared__ float red[256];
  const int n = blockIdx.x;
  const float4* row = (const float4*)(B + (size_t)n * NE);
  float s = 0.f;
  for (int i = threadIdx.x; i < NE / 4; i += 256) {
    float4 v = row[i];
    s += v.x + v.y + v.z + v.w;
  }
  red[threadIdx.x] = s;
  __syncthreads();
  for (int off = 128; off > 0; off >>= 1) {
    if (threadIdx.x < off) red[threadIdx.x] += red[threadIdx.x + off];
    __syncthreads();
  }
  if (threadIdx.x == 0) deg[n] = red[0];
}

// ---------------- build augmented X: [x0 | 1 | deg | 0-pad] ----------------
__global__ __launch_bounds__(256) void k_xaug(const float* __restrict__ x0,
                                              const float* __restrict__ deg,
                                              float* __restrict__ xa) {
  int idx = blockIdx.x * 256 + threadIdx.x;
  if (idx >= NN * WAUG) return;
  int n = idx / WAUG, c = idx % WAUG;
  float v;
  if (c < C)           v = x0[n * C + c];
  else if (c == C)     v = 1.0f;        // -> colsum(B)
  else if (c == C + 1) v = deg[n];      // -> deg @ B
  else                 v = 0.0f;
  xa[idx] = v;
}

// ---------------- pass 1: X1aug = B^T @ Xaug, epilogue scales Y --------------
// LDS slab is stored TRANSPOSED: slabT[c * KTP + k], double buffered, filled by
// async global->LDS loads (per-lane LDS destinations do the transpose).
__global__ __launch_bounds__(128) void k_x1(const float* __restrict__ Binc,
                                            const float* __restrict__ Xaug,
                                            float* __restrict__ x1_out,
                                            float* __restrict__ Y) {
  __shared__ __align__(16) float slabT[2][WAUG * KTP];
  __shared__ float s_cs[4][16];
  __shared__ float s_ws[4][16];

  const int tid  = threadIdx.x;
  const int wave = tid >> 5;
  const int lane = tid & 31;
  const int lm   = lane & 15;
  const int half = lane >> 4;
  const int e0   = blockIdx.x * 64 + wave * 16;

  constexpr int EPT = (KT * WAUG) / 128;  // 20 async dwords per thread per slab
  const uint32_t buf_stride = (uint32_t)(WAUG * KTP * 4);
  const uint32_t base0 = (uint32_t)(uintptr_t)&slabT[0][0];

  uint32_t loff[EPT];   // transposed LDS byte offsets (buffer 0)
  uint32_t goff[EPT];   // global byte offsets within slab
#pragma unroll
  for (int i = 0; i < EPT; ++i) {
    uint32_t s = (uint32_t)tid + 128u * i;
    uint32_t k = s / WAUG, c = s % WAUG;
    loff[i] = base0 + (c * KTP + k) * 4u;
    goff[i] = s * 4u;
  }

  // prologue: fill buffer 0 with slab k0=0
#pragma unroll
  for (int i = 0; i < EPT; ++i) async_ld_b32(loff[i], goff[i], Xaug);

  v8f acc[5];
  const v8f vz = {0.f, 0.f, 0.f, 0.f, 0.f, 0.f, 0.f, 0.f};
#pragma unroll
  for (int t = 0; t < 5; ++t) acc[t] = vz;

  for (int k0 = 0; k0 < NN; k0 += KT) {
    const int buf = (k0 >> 5) & 1;
    if (k0 + KT < NN) {  // prefetch next slab into other buffer
      const float* nsrc = Xaug + (size_t)(k0 + KT) * WAUG;
      const uint32_t boff = (buf ^ 1) ? buf_stride : 0u;
#pragma unroll
      for (int i = 0; i < EPT; ++i) async_ld_b32(loff[i] + boff, goff[i], nsrc);
      wait_asynccnt<EPT>();   // current slab's loads (issued earlier) are done
    } else {
      wait_asynccnt<0>();
    }
    __syncthreads();

    // preload all 8 A fragments of this slab (one vmem clause)
    v2f a[KT / 4];
#pragma unroll
    for (int kk = 0; kk < KT; kk += 4) {
      const float* ap = Binc + (size_t)(k0 + kk + 2 * half) * NE + (e0 + lm);
      a[kk / 4].x = ap[0];
      a[kk / 4].y = ap[NE];
    }

    // per-tile LDS column bases hoisted: kk folds into DS offset immediates
    const float* sb = &slabT[buf][0];
    const float* cb[5];
#pragma unroll
    for (int t = 0; t < 5; ++t) cb[t] = sb + (t * 16 + lm) * KTP + 2 * half;

#pragma unroll
    for (int kk = 0; kk < KT; kk += 4) {
#pragma unroll
      for (int t = 0; t < 5; ++t) {
        v2f b;
        b.x = cb[t][kk];
        b.y = cb[t][kk + 1];
        acc[t] = wmma4(a[kk / 4], b, acc[t]);
      }
    }
    __syncthreads();  // protect buf before it is refilled two slabs later
  }

  // tile 4 holds col 64 = colsum (lm==0 lanes), col 65 = wsum (lm==1 lanes)
#pragma unroll
  for (int r = 0; r < 8; ++r) {
    if (lm == 0) s_cs[wave][r + 8 * half] = acc[4][r];
    if (lm == 1) s_ws[wave][r + 8 * half] = acc[4][r];
  }
  __syncthreads();

  float scale[8];
#pragma unroll
  for (int r = 0; r < 8; ++r) {
    int row = r + 8 * half;
    // edge_degree = wsum/colsum ; scale = rsqrt(edge_degree)
    scale[r] = rsqrtf(s_ws[wave][row] / s_cs[wave][row]);
  }

#pragma unroll
  for (int t = 0; t < 4; ++t) {
#pragma unroll
    for (int r = 0; r < 8; ++r) {
      int e = e0 + r + 8 * half;
      int c = t * 16 + lm;
      float v = acc[t][r];
      x1_out[(size_t)e * C + c] = v;        // raw x_1 (2nd output)
      Y[(size_t)e * C + c] = v * scale[r];  // scaled for pass 2
    }
  }
}

// ---------------- pass 2: msg = B @ Y, fused epilogue (skip + W-GEMM) --------
__global__ __launch_bounds__(128) void k_out(const float* __restrict__ Binc,
                                             const float* __restrict__ Y,
                                             const float* __restrict__ deg,
                                             const float* __restrict__ x0,
                                             const float* __restrict__ W,
                                             float* __restrict__ xout) {
  __shared__ __align__(16) float slabT[2][C * KTP];
  __shared__ __align__(16) float comb[4][16 * C];

  const int tid  = threadIdx.x;
  const int wave = tid >> 5;
  const int lane = tid & 31;
  const int lm   = lane & 15;
  const int half = lane >> 4;
  const int n0   = blockIdx.x * 64 + wave * 16;

  constexpr int EPT = (KT * C) / 128;  // 16 async dwords per thread per slab
  const uint32_t buf_stride = (uint32_t)(C * KTP * 4);
  const uint32_t base0 = (uint32_t)(uintptr_t)&slabT[0][0];

  uint32_t loff[EPT];
  uint32_t goff[EPT];
#pragma unroll
  for (int i = 0; i < EPT; ++i) {
    uint32_t s = (uint32_t)tid + 128u * i;
    uint32_t k = s >> 6, c = s & 63u;
    loff[i] = base0 + (c * KTP + k) * 4u;
    goff[i] = s * 4u;
  }

#pragma unroll
  for (int i = 0; i < EPT; ++i) async_ld_b32(loff[i], goff[i], Y);

  v8f acc[4];
  const v8f vz = {0.f, 0.f, 0.f, 0.f, 0.f, 0.f, 0.f, 0.f};
#pragma unroll
  for (int t = 0; t < 4; ++t) acc[t] = vz;

  for (int k0 = 0; k0 < NE; k0 += KT) {
    const int buf = (k0 >> 5) & 1;
    if (k0 + KT < NE) {
      const float* nsrc = Y + (size_t)(k0 + KT) * C;
      const uint32_t boff = (buf ^ 1) ? buf_stride : 0u;
#pragma unroll
      for (int i = 0; i < EPT; ++i) async_ld_b32(loff[i] + boff, goff[i], nsrc);
      wait_asynccnt<EPT>();
    } else {
      wait_asynccnt<0>();
    }
    __syncthreads();

    // preload all 8 A fragments (contiguous K pair -> b64 loads, one clause)
    v2f a[KT / 4];
    const float* arow = Binc + (size_t)(n0 + lm) * NE + k0 + 2 * half;
#pragma unroll
    for (int kk = 0; kk < KT; kk += 4) a[kk / 4] = *(const v2f*)(arow + kk);

    const float* sb = &slabT[buf][0];
    const float* cb[4];
#pragma unroll
    for (int t = 0; t < 4; ++t) cb[t] = sb + (t * 16 + lm) * KTP + 2 * half;

#pragma unroll
    for (int kk = 0; kk < KT; kk += 4) {
#pragma unroll
      for (int t = 0; t < 4; ++t) {
        v2f b;
        b.x = cb[t][kk];
        b.y = cb[t][kk + 1];
        acc[t] = wmma4(a[kk / 4], b, acc[t]);
      }
    }
    __syncthreads();
  }

  // x_comb = 0.9 * msg * rsqrt(deg) + 0.1 * x0 ; keep in acc and stage in LDS
#pragma unroll
  for (int t = 0; t < 4; ++t) {
#pragma unroll
    for (int r = 0; r < 8; ++r) {
      int row = r + 8 * half;
      int n = n0 + row;
      int c = t * 16 + lm;
      float rd = rsqrtf(deg[n]);
      float xc = 0.9f * (acc[t][r] * rd) + 0.1f * x0[(size_t)n * C + c];
      acc[t][r] = xc;
      comb[wave][row * C + c] = xc;
    }
  }
  __syncthreads();

  // wacc = comb_tile @ W^T  (M=16, N=64, K=64 via WMMA; A re-fragmented in LDS)
  v8f wacc[4];
#pragma unroll
  for (int t = 0; t < 4; ++t) wacc[t] = vz;
  {
    const float* ab = &comb[wave][lm * C + 2 * half];
    const float* wb[4];
#pragma unroll
    for (int t = 0; t < 4; ++t) wb[t] = W + (t * 16 + lm) * C + 2 * half;
#pragma unroll
    for (int k = 0; k < C; k += 4) {
      v2f a;
      a.x = ab[k];
      a.y = ab[k + 1];
#pragma unroll
      for (int t = 0; t < 4; ++t) {
        v2f b;  // Bm[j, c] = W[c, j]
        b.x = wb[t][k];
        b.y = wb[t][k + 1];
        wacc[t] = wmma4(a, b, wacc[t]);
      }
    }
  }

  // x_out = 0.5 * x_comb + 0.5 * (x_comb @ W^T)
#pragma unroll
  for (int t = 0; t < 4; ++t) {
#pragma unroll
    for (int r = 0; r < 8; ++r) {
      int n = n0 + r + 8 * half;
      int c = t * 16 + lm;
      xout[(size_t)n * C + c] = 0.5f * acc[t][r] + 0.5f * wacc[t][r];
    }
  }
}

extern "C" void kernel_launch(void* const* d_in, const int* in_sizes, int n_in,
                              void* d_out, int out_size, void* d_ws,
                              size_t ws_size, hipStream_t stream) {
  const float* x0   = (const float*)d_in[0];
  const float* Binc = (const float*)d_in[1];
  const float* W    = (const float*)d_in[2];

  float* out  = (float*)d_out;
  float* xout = out;                        // [N, C]  (first return)
  float* x1   = out + (size_t)NN * C;       // [E, C]  (second return)

  float* ws   = (float*)d_ws;
  float* deg  = ws;                         // [N]
  float* xaug = ws + NN;                    // [N, 80]
  float* Y    = xaug + (size_t)NN * WAUG;   // [E, C]

  k_degree<<<NN, 256, 0, stream>>>(Binc, deg);
  k_xaug<<<(NN * WAUG + 255) / 256, 256, 0, stream>>>(x0, deg, xaug);
  k_x1<<<NE / 64, 128, 0, stream>>>(Binc, xaug, x1, Y);
  k_out<<<NN / 64, 128, 0, stream>>>(Binc, Y, deg, x0, W, xout);
}